// MaskCNN_65876208386322
// MI455X (gfx1250) — compile-verified
//
#include <hip/hip_runtime.h>

typedef float v2f __attribute__((ext_vector_type(2)));
typedef float v8f __attribute__((ext_vector_type(8)));

#define Bn   32
#define Hin  80
#define Tin  1600
#define C1n  32
#define C2n  32
#define H1n  40
#define T1n  800
#define H2n  20
#define T2n  400

// padded intermediate: [b][c1][HP][TP], halo of 1 on every side
#define HP   42
#define TP   802
#define CSTRIDE (HP * TP)          // 33684 floats per (b,c1) slab

// A-matrix in LDS: K padded 9 -> 12 per c1 (K total 384), row stride 388
#define KPAD 384
#define LDSROW 388                 // 388 % 64 == 4 -> conflict-free fragment reads

// ---------------------------------------------------------------------------
// Kernel 0: zero the halo border of the padded intermediate.
// 1684 halo cells per (b,c1) slab * 1024 slabs = 1,724,416 threads.
// ---------------------------------------------------------------------------
__global__ __launch_bounds__(256) void halo_zero(float* __restrict__ x1p)
{
    int gid = blockIdx.x * 256 + threadIdx.x;
    int bc  = gid / 1684;
    int i   = gid - bc * 1684;
    int h, w;
    if (i < 1604) {                       // rows 0 and 41, full width
        h = (i < 802) ? 0 : (HP - 1);
        w = (i < 802) ? i : i - 802;
    } else {                              // cols 0 and 801, rows 1..40
        int j = i - 1604;
        h = 1 + (j >> 1);
        w = (j & 1) ? (TP - 1) : 0;
    }
    x1p[(long)bc * CSTRIDE + h * TP + w] = 0.f;
}

// ---------------------------------------------------------------------------
// Kernel 1: conv1 (1->32ch, 3x3, s2, p1) + bias + mask + relu -> padded x1.
// One thread per (b, h1, t1); all 32 output channels share one 3x3 patch.
// Also emits the tuple's second output (new lengths L2) from threads 0..31.
// ---------------------------------------------------------------------------
__global__ __launch_bounds__(256) void conv1_mask_relu(
    const float* __restrict__ in, const int* __restrict__ seqlen,
    const float* __restrict__ W1, const float* __restrict__ b1,
    float* __restrict__ x1p, int* __restrict__ lenOut)
{
    __shared__ float w[C1n * 9];
    __shared__ float bb[C1n];
    int tid = threadIdx.x;
    for (int i = tid; i < C1n * 9; i += blockDim.x) w[i] = W1[i];
    if (tid < C1n) bb[tid] = b1[tid];
    __syncthreads();

    long gid = (long)blockIdx.x * blockDim.x + tid;

    if (lenOut != nullptr && gid < Bn) {
        int L   = seqlen[gid];
        int L1v = (L - 1) / 2 + 1;
        lenOut[gid] = (L1v - 1) / 2 + 1;
    }

    int t1 = (int)(gid % T1n);
    int h1 = (int)((gid / T1n) % H1n);
    int b  = (int)(gid / ((long)T1n * H1n));
    if (b >= Bn) return;

    // 3x3 input patch. Ranges: h = 2*h1+kh-1 in [-1,79]; t = 2*t1+kw-1 in [-1,1599]
    // => only kh==0@h1==0 and kw==0@t1==0 can go out of bounds.
    float p[9];
#pragma unroll
    for (int kh = 0; kh < 3; ++kh) {
        int  h   = 2 * h1 + kh - 1;
        bool hok = (kh > 0) || (h1 > 0);
#pragma unroll
        for (int kw = 0; kw < 3; ++kw) {
            int  t   = 2 * t1 + kw - 1;
            bool tok = (kw > 0) || (t1 > 0);
            float v = 0.f;
            if (hok && tok) v = in[((long)b * Hin + h) * Tin + t];
            p[kh * 3 + kw] = v;
        }
    }

    int  L     = seqlen[b];
    int  L1v   = (L - 1) / 2 + 1;
    bool valid = (t1 < L1v);

    // interior of padded layout: (h1+1, t1+1)
    float* dst = x1p + (((long)b * C1n) * HP + (h1 + 1)) * TP + (t1 + 1);
#pragma unroll 4
    for (int c = 0; c < C1n; ++c) {
        float acc = bb[c];
#pragma unroll
        for (int j = 0; j < 9; ++j) acc = fmaf(w[c * 9 + j], p[j], acc);
        acc = valid ? fmaxf(acc, 0.f) : 0.f;
        dst[(long)c * CSTRIDE] = acc;
    }
}

// ---------------------------------------------------------------------------
// Kernel 2: conv2 (32->32ch, 3x3, s2, p1) as implicit GEMM on the FP32 WMMA
// path (V_WMMA_F32_16X16X4_F32). Per wave: M=32 (two 16-row c2 tiles),
// N=16 consecutive t2, K=32 c1 * 12 (9 taps zero-padded to 12) = 384.
// Per c1: 6 unconditional global loads (3x3 patch), 6 LDS A-fragment loads,
// 5 cndmask selects, 6 WMMAs. No divisions, no exec divergence.
//   A 16x4 fp32: lane<16 M=lane (K+0,K+1); lane>=16 M=lane-16 (K+2,K+3)
//   B 4x16 fp32: per lane {val(K+2*hi), val(K+2*hi+1)}, N=lane&15
//   C/D 16x16: VGPR v -> M=v (lanes 0-15) / M=v+8 (lanes 16-31), N=lane&15
// ---------------------------------------------------------------------------
__global__ __launch_bounds__(256) void conv2_wmma(
    const float* __restrict__ x1p, const int* __restrict__ seqlen,
    const float* __restrict__ W2, const float* __restrict__ b2,
    float* __restrict__ out)
{
    __shared__ float w[C2n * LDSROW];
    __shared__ float bb[C2n];
    int tid = threadIdx.x;
    // stage W2 zero-padded: w[c2][c1*12 + r] = (r<9) ? W2[c2][c1*9+r] : 0
    for (int i = tid; i < C2n * KPAD; i += blockDim.x) {
        int c2  = i / KPAD;
        int rem = i - c2 * KPAD;
        int c1  = rem / 12;
        int r   = rem - c1 * 12;
        w[c2 * LDSROW + rem] = (r < 9) ? W2[(c2 * C1n + c1) * 9 + r] : 0.f;
    }
    if (tid < C2n) bb[tid] = b2[tid];
    __syncthreads();

    int wave = tid >> 5;
    int lane = tid & 31;
    int n    = lane & 15;   // N index (t2 within tile)
    int hi   = lane >> 4;   // half-wave selector

    int tile   = blockIdx.x * 8 + wave;   // 16000 tiles = 32*20*25
    int t2base = (tile % 25) * 16;
    int bh     = tile / 25;
    int h2     = bh % H2n;
    int b      = bh / H2n;

    int t2 = t2base + n;

    v8f acc0 = {};
    v8f acc1 = {};

    const float* wr0 = &w[n * LDSROW];          // A rows c2 = 0..15
    const float* wr1 = &w[(n + 16) * LDSROW];   // A rows c2 = 16..31

    // per-lane patch base in padded coords: (2*h2, 2*t2); +1 interior and -1
    // conv offset cancel. All 9 taps are immediate offsets from px.
    const float* px = x1p + (((long)b * C1n) * HP + 2 * h2) * TP + 2 * t2;

#pragma unroll 2
    for (int c1 = 0; c1 < C1n; ++c1) {
        __builtin_prefetch(px + CSTRIDE, 0, 1);   // next c1 slab

        // 3x3 patch: rows are contiguous triples (8B-aligned b64 + b32)
        v2f   r0 = *(const v2f*)(px);             // p0,p1  (kh=0)
        float p2 = px[2];
        v2f   r1 = *(const v2f*)(px + TP);        // p3,p4  (kh=1)
        float p5 = px[TP + 2];
        v2f   r2 = *(const v2f*)(px + 2 * TP);    // p6,p7  (kh=2)
        float p8 = px[2 * TP + 2];

        int kb = c1 * 12 + 2 * hi;

        // ---- K-step 0: taps r = {0,1 | 2,3} ----
        v2f a0 = *(const v2f*)&wr0[kb];
        v2f a1 = *(const v2f*)&wr1[kb];
        v2f bv;
        bv.x = hi ? p2   : r0.x;
        bv.y = hi ? r1.x : r0.y;
        acc0 = __builtin_amdgcn_wmma_f32_16x16x4_f32(false, a0, false, bv, (short)0, acc0, false, false);
        acc1 = __builtin_amdgcn_wmma_f32_16x16x4_f32(false, a1, false, bv, (short)0, acc1, false, false);

        // ---- K-step 1: taps r = {4,5 | 6,7} ----
        a0 = *(const v2f*)&wr0[kb + 4];
        a1 = *(const v2f*)&wr1[kb + 4];
        bv.x = hi ? r2.x : r1.y;
        bv.y = hi ? r2.y : p5;
        acc0 = __builtin_amdgcn_wmma_f32_16x16x4_f32(false, a0, false, bv, (short)0, acc0, false, false);
        acc1 = __builtin_amdgcn_wmma_f32_16x16x4_f32(false, a1, false, bv, (short)0, acc1, false, false);

        // ---- K-step 2: taps r = {8,pad | pad,pad} ----
        a0 = *(const v2f*)&wr0[kb + 8];
        a1 = *(const v2f*)&wr1[kb + 8];
        bv.x = hi ? 0.f : p8;
        bv.y = 0.f;
        acc0 = __builtin_amdgcn_wmma_f32_16x16x4_f32(false, a0, false, bv, (short)0, acc0, false, false);
        acc1 = __builtin_amdgcn_wmma_f32_16x16x4_f32(false, a1, false, bv, (short)0, acc1, false, false);

        px += CSTRIDE;
    }

    int  L     = seqlen[b];
    int  L1v   = (L - 1) / 2 + 1;
    int  L2v   = (L1v - 1) / 2 + 1;
    bool valid = (t2 < L2v);

#pragma unroll
    for (int v = 0; v < 8; ++v) {
        int c2a = v + hi * 8;       // M-tile 0: c2 0..15
        int c2b = c2a + 16;         // M-tile 1: c2 16..31
        float va = acc0[v] + bb[c2a];
        float vb = acc1[v] + bb[c2b];
        va = valid ? fmaxf(va, 0.f) : 0.f;
        vb = valid ? fmaxf(vb, 0.f) : 0.f;
        out[(((long)b * C2n + c2a) * H2n + h2) * T2n + t2] = va;
        out[(((long)b * C2n + c2b) * H2n + h2) * T2n + t2] = vb;
    }
}

// ---------------------------------------------------------------------------
extern "C" void kernel_launch(void* const* d_in, const int* in_sizes, int n_in,
                              void* d_out, int out_size, void* d_ws, size_t ws_size,
                              hipStream_t stream)
{
    const float* in  = (const float*)d_in[0];
    const int*   sl  = (const int*)  d_in[1];
    const float* W1  = (const float*)d_in[2];
    const float* b1  = (const float*)d_in[3];
    const float* W2  = (const float*)d_in[4];
    const float* b2  = (const float*)d_in[5];

    float* out = (float*)d_out;
    const long xElems = (long)Bn * C2n * H2n * T2n;            // 8,192,000
    int* lenOut = (out_size >= (int)(xElems + Bn))
                      ? (int*)(out + xElems) : nullptr;        // tuple output 2: L2

    float* x1p = (float*)d_ws;  // padded intermediate: 32*32*42*802 floats (~138 MB)

    // halo: 1,724,416 cells = 6736 * 256
    halo_zero<<<6736, 256, 0, stream>>>(x1p);
    // conv1: 32*40*800 = 1,024,000 threads
    conv1_mask_relu<<<4000, 256, 0, stream>>>(in, sl, W1, b1, x1p, lenOut);
    // conv2: 16,000 wave-tiles, 8 waves/block
    conv2_wmma<<<2000, 256, 0, stream>>>(x1p, sl, W2, b2, out);
}